// BiLinearInteractionLayer_32384053412590
// MI455X (gfx1250) — compile-verified
//
#include <hip/hip_runtime.h>

// BiLinearInteractionLayer: out[b,p,o] = (sum_d x[b,pi[p],d] * W[p,o,d]) * x[b,pj[p],o]
// B=4096, 32 fields, D=64, 496 pairs. fp32 throughout (matches reference precision).
//
// Strategy (MI455X / gfx1250, wave32):
//  - 496 GEMMs M=4096,N=64,K=64 via V_WMMA_F32_16X16X4_F32 (fp32 in/out).
//  - One pair per blockIdx.y; 128 batch rows per blockIdx.x; 8 waves/block,
//    each wave computes a 16x64 output tile (4 n-tiles x 16 k-steps of WMMA).
//  - W_p (64x64, 16KB) staged into LDS once per block, row stride padded to 68
//    floats -> conflict-free float2 fragment reads (ds_load_2addr_b64 pairs).
//  - A fragments are native-layout float2 global loads (x is L2-resident).
//  - xj values for the elementwise scale are PRELOADED before the WMMA chains
//    so the 64-WMMA stream hides their latency; epilogue is pure mul+store.
//  - Output stores are two contiguous 64B segments per half-wave -> saturates
//    the 520MB HBM write stream, which is the roofline limiter (~22us @23.3TB/s).

#define NUM_FIELDS 32
#define DIM        64
#define BATCH      4096
#define NPAIRS     496
#define ROWS_PER_BLOCK 128
#define LDS_STRIDE 68   // 64 + 4 pad floats: bank-conflict-free B-fragment reads

typedef __attribute__((ext_vector_type(2))) float v2f;
typedef __attribute__((ext_vector_type(8))) float v8f;

__global__ __launch_bounds__(256)
void bilinear_wmma_f32_kernel(const float* __restrict__ x,
                              const float* __restrict__ W,
                              float* __restrict__ out) {
  __shared__ float Wlds[DIM * LDS_STRIDE];

  const int p = blockIdx.y;
  // Decode pair p -> (fi, fj), itertools.combinations(range(32), 2) order.
  int fi = 0, rem = p;
  while (rem >= 31 - fi) { rem -= 31 - fi; ++fi; }
  const int fj = fi + 1 + rem;

  const int tid = threadIdx.x;

  // ---- Stage W_p (64x64 f32) into padded LDS: 1024 float4s over 256 threads.
  const float* Wp = W + (size_t)p * (DIM * DIM);
  #pragma unroll
  for (int it = 0; it < 4; ++it) {
    const int idx = tid + 256 * it;        // float4 index 0..1023
    const int o   = idx >> 4;              // 16 float4 per 64-float row
    const int c   = (idx & 15) << 2;
    const float4 v = ((const float4*)Wp)[idx];
    *(float4*)&Wlds[o * LDS_STRIDE + c] = v;
  }
  __syncthreads();

  const int wave = tid >> 5;
  const int lane = tid & 31;
  const int l15  = lane & 15;
  const int hi   = lane >> 4;              // 0: lanes 0-15, 1: lanes 16-31

  const int row0 = blockIdx.x * ROWS_PER_BLOCK + wave * 16;

  // ---- Preload A fragments: lane (l15,hi) holds {xi[row0+l15][4s+2hi], +1}.
  const float* xi = x + (size_t)(row0 + l15) * (NUM_FIELDS * DIM)
                      + fi * DIM + 2 * hi;
  v2f a[16];
  #pragma unroll
  for (int s = 0; s < 16; ++s)
    a[s] = *(const v2f*)(xi + 4 * s);

  // ---- Preload xj for the epilogue scale (issued before WMMA chains so the
  //      matrix stream hides the latency). Element (t,r) pairs with acc[r] of
  //      n-tile t: row = row0 + r + 8*hi, col = 16*t + l15. Coalesced: two
  //      contiguous 64B segments per half-wave at each (t,r).
  float xjv[32];
  #pragma unroll
  for (int t = 0; t < 4; ++t)
    #pragma unroll
    for (int r = 0; r < 8; ++r)
      xjv[t * 8 + r] = x[(size_t)(row0 + r + 8 * hi) * (NUM_FIELDS * DIM)
                         + fj * DIM + 16 * t + l15];

  // ---- 4 n-tiles of 16x16, each accumulated over K=64 in 16 WMMA steps.
  #pragma unroll
  for (int t = 0; t < 4; ++t) {
    v8f acc = {};
    const float* wb = &Wlds[(16 * t + l15) * LDS_STRIDE + 2 * hi];
    #pragma unroll
    for (int s = 0; s < 16; ++s) {
      const v2f b = *(const v2f*)(wb + 4 * s);
      // D = A(16x4) * B(4x16) + C, fp32. 8 args: neg_a,A,neg_b,B,c_mod,C,reuse_a,reuse_b
      acc = __builtin_amdgcn_wmma_f32_16x16x4_f32(
          false, a[s], false, b, (short)0, acc, false, false);
    }
    // ---- Scale by preloaded xj and store (pure mul+store epilogue).
    #pragma unroll
    for (int r = 0; r < 8; ++r) {
      const int row = row0 + r + 8 * hi;
      const int col = 16 * t + l15;
      out[((size_t)row * NPAIRS + p) * DIM + col] = acc[r] * xjv[t * 8 + r];
    }
  }
}

extern "C" void kernel_launch(void* const* d_in, const int* in_sizes, int n_in,
                              void* d_out, int out_size, void* d_ws, size_t ws_size,
                              hipStream_t stream) {
  (void)in_sizes; (void)n_in; (void)out_size; (void)d_ws; (void)ws_size;
  const float* x = (const float*)d_in[0];   // (4096, 32, 64) f32
  const float* W = (const float*)d_in[1];   // (496, 64, 64) f32
  float* out = (float*)d_out;               // (4096, 496, 64) f32

  dim3 grid(BATCH / ROWS_PER_BLOCK, NPAIRS);  // (32, 496)
  bilinear_wmma_f32_kernel<<<grid, 256, 0, stream>>>(x, W, out);
}